// DistMult_KBLN_79164837200204
// MI455X (gfx1250) — compile-verified
//
#include <hip/hip_runtime.h>

// Problem constants
#define B_SZ   64
#define DIM    200
#define DIMP   204      // padded LDS row (204 % 64 = 12 -> conflict-free strided reads)
#define NLIT   100
#define NENT   40000
#define NREL   1345
#define TN     32       // entities per workgroup (NENT % TN == 0 -> 1250 blocks)

typedef __attribute__((ext_vector_type(2))) float v2f;
typedef __attribute__((ext_vector_type(8))) float v8f;

// v_exp_f32 (base-2 exponential) — direct hardware transcendental
__device__ __forceinline__ float fast_exp2(float x) {
    return __builtin_amdgcn_exp2f(x);
}

// ---------------------------------------------------------------------------
// Prep kernel: one block per batch row b. Gathers embedding rows once so the
// main kernel never touches the gather-indexed paths.
//   wsA[b][d]    = E[e1[b]][d] * R[r[b]][d]
//   wsAHat[b][l] = num_lit[e1[b]][l] - c[l]
//   wsW[b][l]    = nf_weights[r[b]][l]
//   wsScale[l]   = -log2(e)/var[l]          (so exp(-x^2/var) = exp2(x^2*scale))
// ---------------------------------------------------------------------------
__global__ __launch_bounds__(256)
void distmult_prep(const int* __restrict__ e1_idx, const int* __restrict__ r_idx,
                   const float* __restrict__ E, const float* __restrict__ R,
                   const float* __restrict__ num_lit, const float* __restrict__ c,
                   const float* __restrict__ var, const float* __restrict__ nf,
                   float* __restrict__ wsA, float* __restrict__ wsAHat,
                   float* __restrict__ wsW, float* __restrict__ wsScale)
{
    const int b  = blockIdx.x;
    const int e1 = e1_idx[b];
    const int r  = r_idx[b];

    for (int d = threadIdx.x; d < DIM; d += blockDim.x)
        wsA[b * DIM + d] = E[(size_t)e1 * DIM + d] * R[(size_t)r * DIM + d];

    for (int l = threadIdx.x; l < NLIT; l += blockDim.x) {
        wsAHat[b * NLIT + l] = num_lit[(size_t)e1 * NLIT + l] - c[l];
        wsW[b * NLIT + l]    = nf[(size_t)r * NLIT + l];
        if (b == 0)
            wsScale[l] = -1.44269504088896340736f / var[l];
    }
}

// ---------------------------------------------------------------------------
// Main kernel: 256 threads = 8 wave32.
// Wave w: rb = w>>1 (M block of 16 rows), cb = w&1 (N block of 16 entities).
// Workgroup covers all 64 b-rows x 32 entities.
// ---------------------------------------------------------------------------
__global__ __launch_bounds__(256)
void distmult_main(const float* __restrict__ E, const float* __restrict__ num_lit,
                   const float* __restrict__ wsA, const float* __restrict__ wsAHat,
                   const float* __restrict__ wsW, const float* __restrict__ wsScale,
                   float* __restrict__ out)
{
    __shared__ float sA[B_SZ][DIMP];     // (e1*r), padded rows
    __shared__ float sE[TN][DIMP];       // E_weight tile, padded rows
    __shared__ float sAHat[B_SZ][NLIT];  // n_h - c
    __shared__ float sW[B_SZ][NLIT];     // nf weights per b
    __shared__ float sLit[TN][NLIT];     // num_lit tile
    __shared__ float sScale[NLIT];       // -log2e/var

    const int tid   = threadIdx.x;
    const int eBase = blockIdx.x * TN;

    // ---- Phase 1: stage LDS (coalesced) ----
    for (int i = tid; i < B_SZ * DIM; i += 256) {
        int b = i / DIM, d = i - b * DIM;
        sA[b][d] = wsA[i];
    }
    for (int i = tid; i < TN * DIM; i += 256) {
        int n = i / DIM, d = i - n * DIM;
        sE[n][d] = E[(size_t)(eBase + n) * DIM + d];
    }
    for (int i = tid; i < B_SZ * NLIT; i += 256) {
        (&sAHat[0][0])[i] = wsAHat[i];
        (&sW[0][0])[i]    = wsW[i];
    }
    for (int i = tid; i < TN * NLIT; i += 256)
        (&sLit[0][0])[i] = num_lit[(size_t)eBase * NLIT + i];
    if (tid < NLIT)
        sScale[tid] = wsScale[tid];
    __syncthreads();

    // ---- Wave/lane geometry ----
    const int lane = tid & 31;
    const int wave = tid >> 5;
    const int rb   = wave >> 1;          // 0..3 : 16-row block of b
    const int cb   = wave & 1;           // 0..1 : 16-col block of entities
    const int lo   = lane & 15;
    const int hi   = lane >> 4;          // 0 or 1
    const int mA   = rb * 16 + lo;       // A-matrix row held by this lane
    const int nB   = cb * 16 + lo;       // B-matrix col held by this lane
    const int kOff = hi * 2;             // ISA 16x4 fp32 layout: lanes16-31 hold K=2,3

    // ---- Phase 2: score_l via V_WMMA_F32_16X16X4_F32, K-loop over DIM ----
    v8f acc = {};
    for (int k0 = 0; k0 < DIM; k0 += 4) {
        float2 af = *(const float2*)&sA[mA][k0 + kOff];
        float2 bf = *(const float2*)&sE[nB][k0 + kOff];
        v2f av; av.x = af.x; av.y = af.y;
        v2f bv; bv.x = bf.x; bv.y = bf.y;
        acc = __builtin_amdgcn_wmma_f32_16x16x4_f32(
            false, av, false, bv, (short)0, acc, false, false);
    }

    // ---- Phase 3: score_n (RBF) accumulated in the C/D lane layout ----
    // acc element i sits at (M = rb*16 + i + hi*8, N = eBase + cb*16 + lo)
    float accN[8];
#pragma unroll
    for (int i = 0; i < 8; ++i) accN[i] = 0.0f;

    for (int l4 = 0; l4 < NLIT; l4 += 4) {
        const float4 nl = *(const float4*)&sLit[nB][l4];
        const float4 sc = *(const float4*)&sScale[l4];
#pragma unroll
        for (int i = 0; i < 8; ++i) {
            const int m = rb * 16 + i + hi * 8;
            const float4 ah = *(const float4*)&sAHat[m][l4];
            const float4 ww = *(const float4*)&sW[m][l4];
            float t0 = ah.x - nl.x; accN[i] += ww.x * fast_exp2(t0 * t0 * sc.x);
            float t1 = ah.y - nl.y; accN[i] += ww.y * fast_exp2(t1 * t1 * sc.y);
            float t2 = ah.z - nl.z; accN[i] += ww.z * fast_exp2(t2 * t2 * sc.z);
            float t3 = ah.w - nl.w; accN[i] += ww.w * fast_exp2(t3 * t3 * sc.w);
        }
    }

    // ---- Phase 4: sigmoid + store ----
    const int e = eBase + cb * 16 + lo;
#pragma unroll
    for (int i = 0; i < 8; ++i) {
        const int b = rb * 16 + i + hi * 8;
        const float x = acc[i] + accN[i];
        const float sig = 1.0f / (1.0f + fast_exp2(-x * 1.44269504088896340736f));
        out[(size_t)b * NENT + e] = sig;
    }
}

// ---------------------------------------------------------------------------
// Launch
// ---------------------------------------------------------------------------
extern "C" void kernel_launch(void* const* d_in, const int* in_sizes, int n_in,
                              void* d_out, int out_size, void* d_ws, size_t ws_size,
                              hipStream_t stream) {
    const int*   e1_idx  = (const int*)d_in[0];
    const int*   r_idx   = (const int*)d_in[1];
    const float* E       = (const float*)d_in[2];
    const float* R       = (const float*)d_in[3];
    const float* num_lit = (const float*)d_in[4];
    const float* c       = (const float*)d_in[5];
    const float* var     = (const float*)d_in[6];
    const float* nf      = (const float*)d_in[7];
    float*       out     = (float*)d_out;

    float* ws      = (float*)d_ws;
    float* wsA     = ws;                         // 64*200 floats
    float* wsAHat  = wsA    + B_SZ * DIM;        // 64*100
    float* wsW     = wsAHat + B_SZ * NLIT;       // 64*100
    float* wsScale = wsW    + B_SZ * NLIT;       // 100        (~103 KB total)

    distmult_prep<<<B_SZ, 256, 0, stream>>>(e1_idx, r_idx, E, R, num_lit, c, var, nf,
                                            wsA, wsAHat, wsW, wsScale);
    distmult_main<<<NENT / TN, 256, 0, stream>>>(E, num_lit, wsA, wsAHat, wsW,
                                                 wsScale, out);
}